// DepthWarpingLayer_18073222381996
// MI455X (gfx1250) — compile-verified
//
#include <hip/hip_runtime.h>

// Problem constants (fixed by the reference): B=16, H=1024, W=1280, C=1.
#define BB 16
#define HH 1024
#define WW 1280

// Native clang vector type: required by __builtin_nontemporal_{load,store},
// lowers to global_{load,store}_b128 with TH=NT.
typedef float v4f __attribute__((ext_vector_type(4)));

// ---------------------------------------------------------------------------
// Kernel 1: per-batch coefficient precompute (one lane per batch, one wave).
// Computes, per batch b:
//   Kinv = inverse(intrinsic)            (3x3 adjugate)
//   temp = K @ R^T
//   M    = temp @ Kinv                   (9 floats)
//   Wv   = temp @ (-t)                   (3 floats)
//   M2r2 = row2 of (K @ R) @ Kinv        (3 floats)  -- only row 2 is used
//   W2_2 = row2 of (K @ t)               (1 float)
// Packed as 16 floats per batch into d_ws.
// ---------------------------------------------------------------------------
__global__ void depthwarp_precompute(const float* __restrict__ K9,
                                     const float* __restrict__ Rb,
                                     const float* __restrict__ tb,
                                     float* __restrict__ coef) {
  const int b = threadIdx.x;
  if (b >= BB) return;

  float k[9], r[9], t[3], ki[9], tmp[9];
#pragma unroll
  for (int i = 0; i < 9; ++i) k[i] = K9[i];
#pragma unroll
  for (int i = 0; i < 9; ++i) r[i] = Rb[b * 9 + i];
#pragma unroll
  for (int i = 0; i < 3; ++i) t[i] = tb[b * 3 + i];

  // 3x3 inverse via adjugate.
  const float det = k[0] * (k[4] * k[8] - k[5] * k[7])
                  - k[1] * (k[3] * k[8] - k[5] * k[6])
                  + k[2] * (k[3] * k[7] - k[4] * k[6]);
  const float id = 1.0f / det;
  ki[0] = (k[4] * k[8] - k[5] * k[7]) * id;
  ki[1] = (k[2] * k[7] - k[1] * k[8]) * id;
  ki[2] = (k[1] * k[5] - k[2] * k[4]) * id;
  ki[3] = (k[5] * k[6] - k[3] * k[8]) * id;
  ki[4] = (k[0] * k[8] - k[2] * k[6]) * id;
  ki[5] = (k[2] * k[3] - k[0] * k[5]) * id;
  ki[6] = (k[3] * k[7] - k[4] * k[6]) * id;
  ki[7] = (k[1] * k[6] - k[0] * k[7]) * id;
  ki[8] = (k[0] * k[4] - k[1] * k[3]) * id;

  // temp = K @ R^T  : temp[i][c] = sum_j K[i][j] * R[c][j]
#pragma unroll
  for (int i = 0; i < 3; ++i)
#pragma unroll
    for (int c = 0; c < 3; ++c)
      tmp[3 * i + c] = k[3 * i + 0] * r[3 * c + 0] +
                       k[3 * i + 1] * r[3 * c + 1] +
                       k[3 * i + 2] * r[3 * c + 2];

  float* o = coef + b * 16;

  // M = temp @ Kinv
#pragma unroll
  for (int i = 0; i < 3; ++i)
#pragma unroll
    for (int j = 0; j < 3; ++j)
      o[3 * i + j] = tmp[3 * i + 0] * ki[j] +
                     tmp[3 * i + 1] * ki[3 + j] +
                     tmp[3 * i + 2] * ki[6 + j];

  // Wv = temp @ (-t)
#pragma unroll
  for (int i = 0; i < 3; ++i)
    o[9 + i] = -(tmp[3 * i + 0] * t[0] + tmp[3 * i + 1] * t[1] + tmp[3 * i + 2] * t[2]);

  // kr2 = row2 of K @ R : kr2[c] = sum_j K[2][j] * R[j][c]
  float kr2[3];
#pragma unroll
  for (int c = 0; c < 3; ++c)
    kr2[c] = k[6] * r[c] + k[7] * r[3 + c] + k[8] * r[6 + c];

  // M2r2 = kr2 @ Kinv
#pragma unroll
  for (int j = 0; j < 3; ++j)
    o[12 + j] = kr2[0] * ki[j] + kr2[1] * ki[3 + j] + kr2[2] * ki[6 + j];

  // W2_2 = row2 of K @ t
  o[15] = k[6] * t[0] + k[7] * t[1] + k[8] * t[2];
}

// ---------------------------------------------------------------------------
// Kernel 2: fused warp + bilinear-gather. 4 pixels per thread (row-aligned),
// b128 NT streaming for depth1/out, temporal (L2-resident) gathers of depth2,
// d1_calc fused into the gather taps (no intermediate image in HBM).
// ---------------------------------------------------------------------------
__global__ __launch_bounds__(256)
void depthwarp_main(const float* __restrict__ d1,
                    const float* __restrict__ d2,
                    const float* __restrict__ coef,
                    float* __restrict__ out) {
  const int b = blockIdx.y;                 // uniform -> coef loads scalarize
  const float* c = coef + b * 16;
  const float M00 = c[0],  M01 = c[1],  M02 = c[2];
  const float M10 = c[3],  M11 = c[4],  M12 = c[5];
  const float M20 = c[6],  M21 = c[7],  M22 = c[8];
  const float Wv0 = c[9],  Wv1 = c[10], Wv2 = c[11];
  const float P0  = c[12], P1  = c[13], P2  = c[14];
  const float Q   = c[15];

  const int tile = blockIdx.x * 256 + threadIdx.x; // 4 consecutive pixels
  const int pix  = tile * 4;                       // W % 4 == 0 -> same row
  const int y    = pix / WW;
  const int x    = pix - y * WW;
  const int imgBase = b * (HH * WW);

  // Stream depth1 with non-temporal hint (read-once; keep L2 for gathers).
  const v4f z1v =
      __builtin_nontemporal_load((const v4f*)(d1 + imgBase + pix));
  const float z1a[4] = {z1v.x, z1v.y, z1v.z, z1v.w};
  const float vf = (float)y;

  float res[4];
#pragma unroll
  for (int i = 0; i < 4; ++i) {
    const float uf = (float)(x + i);
    const float z1 = z1a[i];

    const float z2 = fmaf(z1, fmaf(M20, uf, fmaf(M21, vf, M22)), Wv2);
    const float nu = fmaf(z1, fmaf(M00, uf, fmaf(M01, vf, M02)), Wv0);
    const float nv = fmaf(z1, fmaf(M10, uf, fmaf(M11, vf, M12)), Wv1);
    const float u2 = nu / z2;
    const float v2 = nv / z2;

    // Bilinear taps, clip-then-weight exactly as the reference.
    int x0 = (int)floorf(u2);
    int y0 = (int)floorf(v2);
    int x1 = min(max(x0 + 1, 0), WW - 1);
    x0     = min(max(x0,     0), WW - 1);
    int y1 = min(max(y0 + 1, 0), HH - 1);
    y0     = min(max(y0,     0), HH - 1);

    const float x0f = (float)x0, x1f = (float)x1;
    const float y0f = (float)y0, y1f = (float)y1;
    const float wa = (x1f - u2) * (y1f - v2);
    const float wb = (x1f - u2) * (v2 - y0f);
    const float wc = (u2 - x0f) * (y1f - v2);
    const float wd = (u2 - x0f) * (v2 - y0f);

    // Gather depth2 and fuse d1_calc = W2_2 + d2 * (P0*x + P1*y + P2).
    const int r0 = imgBase + y0 * WW;
    const int r1 = imgBase + y1 * WW;
    const float da = d2[r0 + x0];
    const float db = d2[r1 + x0];
    const float dc = d2[r0 + x1];
    const float dd = d2[r1 + x1];

    const float ga = fmaf(da, fmaf(P0, x0f, fmaf(P1, y0f, P2)), Q);
    const float gb = fmaf(db, fmaf(P0, x0f, fmaf(P1, y1f, P2)), Q);
    const float gc = fmaf(dc, fmaf(P0, x1f, fmaf(P1, y0f, P2)), Q);
    const float gd = fmaf(dd, fmaf(P0, x1f, fmaf(P1, y1f, P2)), Q);

    res[i] = wa * ga + wb * gb + wc * gc + wd * gd;
  }

  v4f o4;
  o4.x = res[0]; o4.y = res[1]; o4.z = res[2]; o4.w = res[3];
  __builtin_nontemporal_store(o4, (v4f*)(out + imgBase + pix));
}

// ---------------------------------------------------------------------------
// Launch wrapper.
// Input order (setup_inputs dict): depth_map_1, depth_map_2, translation,
// rotation, intrinsic. Output: [B,H,W,1] f32.
// ---------------------------------------------------------------------------
extern "C" void kernel_launch(void* const* d_in, const int* in_sizes, int n_in,
                              void* d_out, int out_size, void* d_ws, size_t ws_size,
                              hipStream_t stream) {
  const float* depth1    = (const float*)d_in[0];
  const float* depth2    = (const float*)d_in[1];
  const float* trans     = (const float*)d_in[2];
  const float* rot       = (const float*)d_in[3];
  const float* intrinsic = (const float*)d_in[4];
  float* coef = (float*)d_ws;   // 16 batches * 16 floats = 1 KiB
  float* out  = (float*)d_out;

  depthwarp_precompute<<<1, 32, 0, stream>>>(intrinsic, rot, trans, coef);

  // H*W/4 = 327680 tiles; /256 = 1280 blocks per image, exact.
  dim3 grid((HH * WW / 4) / 256, BB);
  dim3 block(256);
  depthwarp_main<<<grid, block, 0, stream>>>(depth1, depth2, coef, out);
}